// attention_41171556499578
// MI455X (gfx1250) — compile-verified
//
#include <hip/hip_runtime.h>
#include <math.h>

// Problem constants (B=2, H=8 folded into BH).
#define NQ 2048
#define DIM 64
#define BH 16
#define NN ((size_t)NQ * (size_t)NQ)          // 4,194,304 = 1<<22
#define TOT ((size_t)BH * NN)                 // 67,108,864 = 1<<26

typedef __attribute__((ext_vector_type(2))) float v2f;
typedef __attribute__((ext_vector_type(8))) float v8f;

// ---------------------------------------------------------------------------
// Kernel 1: S = Q K^T per (b,h) using V_WMMA_F32_16X16X4_F32.
// One wave computes one 16x16 tile of S; block = 4 waves = 16x64 of S.
// A (16x4 f32): lanes 0-15 hold M=0..15 with K={0,1}; lanes 16-31 K={2,3}.
// B (4x16 f32): lanes 0-15 hold N=0..15 with K={0,1}; lanes 16-31 K={2,3}.
// C/D (16x16 f32, 8 VGPRs): VGPR r -> M = r + 8*(lane>=16), N = lane%16.
// Q/K are 17 MB total -> fully L2 resident; S stores stay RT so the stats
// passes can hit them in the 192 MB L2.
// ---------------------------------------------------------------------------
__global__ void qk_wmma_kernel(const float* __restrict__ q,
                               const float* __restrict__ k,
                               float* __restrict__ s) {
    const int lane = threadIdx.x & 31;
    const int wave = threadIdx.x >> 5;
    const int bh = blockIdx.z;
    const int m0 = blockIdx.y * 16;
    const int n0 = (blockIdx.x * 4 + wave) * 16;

    const int half = lane >> 4;   // 0 or 1 (selects K pair {0,1} vs {2,3})
    const int l16  = lane & 15;

    const float* qrow = q + (size_t)bh * NQ * DIM + (size_t)(m0 + l16) * DIM + 2 * half;
    const float* krow = k + (size_t)bh * NQ * DIM + (size_t)(n0 + l16) * DIM + 2 * half;

    v8f acc = {};
#pragma unroll
    for (int k0 = 0; k0 < DIM; k0 += 4) {
        v2f a = *reinterpret_cast<const v2f*>(qrow + k0);  // Q[m][k0+2h .. +1]
        v2f b = *reinterpret_cast<const v2f*>(krow + k0);  // K[n][k0+2h .. +1]
        // 8 args: (neg_a, A, neg_b, B, c_mod, C, reuse_a, reuse_b)
        acc = __builtin_amdgcn_wmma_f32_16x16x4_f32(false, a, false, b,
                                                    (short)0, acc, false, false);
    }

    float* srow = s + (size_t)bh * NN + (size_t)(m0 + 8 * half) * NQ + n0 + l16;
#pragma unroll
    for (int r = 0; r < 8; ++r) {
        srow[(size_t)r * NQ] = acc[r];
    }
}

// ---------------------------------------------------------------------------
// Kernel 2: per-row online max + sum(exp(s-max)) + sum(exp((s-max)/8)).
// One wave per row (wave32); lanes stride the 2048 columns (coalesced 128B).
// Single read pass over S via online (rescaling) accumulation.
// ---------------------------------------------------------------------------
__global__ void row_stats_kernel(const float* __restrict__ s,
                                 float* __restrict__ rowmax,
                                 float* __restrict__ rowsum1,
                                 float* __restrict__ rowsum8) {
    const int lane = threadIdx.x & 31;
    const int wave = threadIdx.x >> 5;
    const int row = blockIdx.x * 8 + wave;          // over BH*NQ rows
    const float* sr = s + (size_t)row * NQ;

    float m = -__builtin_inff(), l1 = 0.f, l8 = 0.f;
    for (int j = lane; j < NQ; j += 32) {
        float v = sr[j];
        if (v > m) {
            float d = m - v;                        // <= 0
            l1 = l1 * __expf(d) + 1.f;
            l8 = l8 * __expf(0.125f * d) + 1.f;
            m = v;
        } else {
            float d = v - m;
            l1 += __expf(d);
            l8 += __expf(0.125f * d);
        }
    }
    // Cross-lane merge of (m, l1, l8) with rescaling.
#pragma unroll
    for (int off = 16; off >= 1; off >>= 1) {
        float m2  = __shfl_xor(m, off, 32);
        float l12 = __shfl_xor(l1, off, 32);
        float l82 = __shfl_xor(l8, off, 32);
        float M = fmaxf(m, m2);
        l1 = l1 * __expf(m - M) + l12 * __expf(m2 - M);
        l8 = l8 * __expf(0.125f * (m - M)) + l82 * __expf(0.125f * (m2 - M));
        m = M;
    }
    if (lane == 0) {
        rowmax[row]  = m;
        rowsum1[row] = l1;
        rowsum8[row] = l8;
    }
}

// ---------------------------------------------------------------------------
// Kernel 3: per-column online max + sum(exp(s-max)) (softmax over axis=2).
// Block = 256 threads = 256 adjacent columns; loop down rows (coalesced 1KB).
// ---------------------------------------------------------------------------
__global__ void col_stats_kernel(const float* __restrict__ s,
                                 float* __restrict__ colmax,
                                 float* __restrict__ colsum) {
    const int bh = blockIdx.y;
    const int c = blockIdx.x * blockDim.x + threadIdx.x;
    const float* sb = s + (size_t)bh * NN + c;

    float m = -__builtin_inff(), l = 0.f;
    for (int r = 0; r < NQ; ++r) {
        float v = sb[(size_t)r * NQ];
        if (v > m) {
            l = l * __expf(m - v) + 1.f;
            m = v;
        } else {
            l += __expf(v - m);
        }
    }
    colmax[bh * NQ + c] = m;
    colsum[bh * NQ + c] = l;
}

// ---------------------------------------------------------------------------
// Kernel 4: finalize. Reads S (staged in out[0..TOT)), writes
//   out0 = scores1 + scores2, out1 = scores1, out2 = scores2.
// scores2 = softmax(s/8, axis=-1); scores1 = softmax(s,axis=2)*softmax(s,axis=3)
// Pure streaming: 256 MB read + 768 MB write.
// The S read is last-use and the 768 MB of output is never re-read, so all
// of them get non-temporal hints to avoid evicting L2-resident S/stat lines.
// ---------------------------------------------------------------------------
__global__ void finalize_kernel(float* __restrict__ out,
                                const float* __restrict__ rowmax,
                                const float* __restrict__ rowsum1,
                                const float* __restrict__ rowsum8,
                                const float* __restrict__ colmax,
                                const float* __restrict__ colsum) {
    const size_t base = (size_t)blockIdx.x * 1024 + threadIdx.x;
#pragma unroll
    for (int it = 0; it < 4; ++it) {
        const size_t i = base + (size_t)it * 256;
        const float sv = __builtin_nontemporal_load(out + i);   // S value (last use)
        const int bh  = (int)(i >> 22);
        const int rem = (int)(i & (NN - 1));
        const int row = rem >> 11;
        const int col = rem & (NQ - 1);
        const int ri = (bh << 11) | row;
        const int ci = (bh << 11) | col;

        const float rm = rowmax[ri];
        const float e1 = __expf(sv - rm);
        const float sc_row = e1 / rowsum1[ri];                       // softmax axis=3
        const float sc_col = __expf(sv - colmax[ci]) / colsum[ci];   // softmax axis=2
        const float scores1 = sc_col * sc_row;
        const float scores2 = __expf(0.125f * (sv - rm)) / rowsum8[ri];

        __builtin_nontemporal_store(scores1 + scores2, out + i);
        __builtin_nontemporal_store(scores1, out + i + TOT);
        __builtin_nontemporal_store(scores2, out + i + 2 * TOT);
    }
}

// ---------------------------------------------------------------------------
extern "C" void kernel_launch(void* const* d_in, const int* in_sizes, int n_in,
                              void* d_out, int out_size, void* d_ws, size_t ws_size,
                              hipStream_t stream) {
    const float* q = (const float*)d_in[0];
    const float* k = (const float*)d_in[1];
    // d_in[2] (v) is unused by the reference math.
    float* out = (float*)d_out;
    float* ws = (float*)d_ws;

    const int BHN = BH * NQ;                 // 32768
    float* rowmax  = ws;
    float* rowsum1 = ws + BHN;
    float* rowsum8 = ws + 2 * BHN;
    float* colmax  = ws + 3 * BHN;
    float* colsum  = ws + 4 * BHN;          // total 5*BHN*4 = 640 KB of ws

    float* S = out;                          // stage S in first output region

    // 1) S = Q K^T  (WMMA fp32)
    dim3 g1(NQ / 64, NQ / 16, BH);           // 32 x 128 x 16 blocks, 4 waves each
    qk_wmma_kernel<<<g1, 128, 0, stream>>>(q, k, S);

    // 2) row statistics (one wave per row, 8 rows per block)
    row_stats_kernel<<<dim3(BHN / 8), 256, 0, stream>>>(S, rowmax, rowsum1, rowsum8);

    // 3) column statistics
    col_stats_kernel<<<dim3(NQ / 256, BH), 256, 0, stream>>>(S, colmax, colsum);

    // 4) finalize all three outputs (non-temporal streaming)
    finalize_kernel<<<dim3((unsigned)(TOT / 1024)), 256, 0, stream>>>(
        out, rowmax, rowsum1, rowsum8, colmax, colsum);
}